// CustomChamferLoss_42795054137808
// MI455X (gfx1250) — compile-verified
//
#include <hip/hip_runtime.h>

// CDNA5 / gfx1250 Chamfer distance via V_WMMA_F32_16X16X4_F32.
//
// dist_ij = ||x_i||^2 + ||y_j||^2 - 2 x_i.y_j   (D=3, K padded to 4)
// Fold:  A[i] = [-2x0, -2x1, -2x2, 1],  B[:,j] = [y0, y1, y2, ||y_j||^2]
//   =>   WMMA out_ij = -2 x.y + ||y_j||^2
// ||x_i||^2 is constant per row -> doesn't affect min over j; added in epilogue.
// Both chamfer directions run in one launch (blockIdx.z selects roles).
// Each wave holds 4 A fragments (64 rows); column loop manually unrolled x2 so
// min updates pair into v_min3_num_f32 (inline asm: avoids LLVM's sNaN
// canonicalize max+min expansion of fminf on WMMA outputs).

typedef float v2f __attribute__((ext_vector_type(2)));
typedef float v8f __attribute__((ext_vector_type(8)));

#define AROWS 4                                  // 16-row A fragments per wave
#define WAVES_PER_BLOCK 8
#define ROWS_PER_BLOCK (AROWS * 16 * WAVES_PER_BLOCK)   // 512

// r = min3(r, d0, d1)  -- single VOP3 v_min3_num_f32
__device__ __forceinline__ void min3_upd(float& r, float d0, float d1)
{
    asm("v_min3_num_f32 %0, %1, %2, %3" : "=v"(r) : "v"(r), "v"(d0), "v"(d1));
}

__global__ __launch_bounds__(256)
void chamfer_kernel(const float* __restrict__ x, const float* __restrict__ y,
                    int N, int M, float* __restrict__ ws)
{
    const int dir = blockIdx.z;                  // 0: x->y   1: y->x
    const float* __restrict__ rowsPtr = dir ? y : x;
    const float* __restrict__ colsPtr = dir ? x : y;
    const int rows = dir ? M : N;
    const int cols = dir ? N : M;
    float* __restrict__ accum = ws + dir;

    const int b      = blockIdx.y;
    const int lane   = threadIdx.x & 31;
    const int waveId = threadIdx.x >> 5;
    const int l16    = lane & 15;
    const int hi     = lane >> 4;                // 0: K=0,1 half   1: K=2,3 half

    const int waveRowBase = blockIdx.x * ROWS_PER_BLOCK + waveId * (AROWS * 16);
    if (waveRowBase >= rows) return;             // wave-uniform guard (EXEC stays full)

    const float* __restrict__ rb = rowsPtr + (size_t)b * (size_t)rows * 3;
    const float* __restrict__ cb = colsPtr + (size_t)b * (size_t)cols * 3;

    // ---- A fragments (16x4 f32): lanes 0-15 carry K0,K1 ; lanes 16-31 carry K2,K3
    v2f   a[AROWS];
    float xn2[AROWS];
#pragma unroll
    for (int f = 0; f < AROWS; ++f) {
        const int r = waveRowBase + f * 16 + l16;
        const float x0 = rb[r * 3 + 0];
        const float x1 = rb[r * 3 + 1];
        const float x2 = rb[r * 3 + 2];
        xn2[f] = x0 * x0 + x1 * x1 + x2 * x2;
        a[f].x = hi ? (-2.0f * x2) : (-2.0f * x0);   // K2 : K0
        a[f].y = hi ? 1.0f         : (-2.0f * x1);   // K3 : K1
    }

    float rmin[AROWS][8];
#pragma unroll
    for (int f = 0; f < AROWS; ++f)
#pragma unroll
        for (int v = 0; v < 8; ++v) rmin[f][v] = 3.402823466e38f;

    // ---- stream column tiles, 2 tiles (32 columns) per iteration ----
    int ct = 0;
    for (; ct + 32 <= cols; ct += 32) {
        const int c0 = ct + l16;
        const int c1 = c0 + 16;

        const float p0 = cb[c0 * 3 + 0];
        const float p1 = cb[c0 * 3 + 1];
        const float p2 = cb[c0 * 3 + 2];
        const float q0 = cb[c1 * 3 + 0];
        const float q1 = cb[c1 * 3 + 1];
        const float q2 = cb[c1 * 3 + 2];
        const float pn2 = p0 * p0 + p1 * p1 + p2 * p2;
        const float qn2 = q0 * q0 + q1 * q1 + q2 * q2;

        // B fragments (4x16 f32): lanes 0-15: K0,K1 rows ; lanes 16-31: K2,K3 rows
        v2f bf0, bf1;
        bf0.x = hi ? p2  : p0;   bf0.y = hi ? pn2 : p1;
        bf1.x = hi ? q2  : q0;   bf1.y = hi ? qn2 : q1;

#pragma unroll
        for (int f = 0; f < AROWS; ++f) {
            v8f cz0 = {}, cz1 = {};
            v8f d0 = __builtin_amdgcn_wmma_f32_16x16x4_f32(
                false, a[f], false, bf0, (short)0, cz0, false, false);
            v8f d1 = __builtin_amdgcn_wmma_f32_16x16x4_f32(
                false, a[f], false, bf1, (short)0, cz1, false, false);
#pragma unroll
            for (int v = 0; v < 8; ++v)
                min3_upd(rmin[f][v], d0[v], d1[v]);
        }
    }
    // remainder (unused when cols % 32 == 0)
    for (; ct < cols; ct += 16) {
        const int c = ct + l16;
        const float p0 = cb[c * 3 + 0];
        const float p1 = cb[c * 3 + 1];
        const float p2 = cb[c * 3 + 2];
        const float pn2 = p0 * p0 + p1 * p1 + p2 * p2;
        v2f bf;
        bf.x = hi ? p2  : p0;
        bf.y = hi ? pn2 : p1;
#pragma unroll
        for (int f = 0; f < AROWS; ++f) {
            v8f cz = {};
            v8f d = __builtin_amdgcn_wmma_f32_16x16x4_f32(
                false, a[f], false, bf, (short)0, cz, false, false);
#pragma unroll
            for (int v = 0; v < 8; ++v)
                rmin[f][v] = fminf(rmin[f][v], d[v]);
        }
    }

    // ---- epilogue: reduce mins across the 16 columns in each half-wave ----
    // C/D layout: vgpr v, lanes 0-15 -> row f*16+v ; lanes 16-31 -> row f*16+v+8.
    float total = 0.0f;
#pragma unroll
    for (int f = 0; f < AROWS; ++f) {
        float s = 0.0f;
#pragma unroll
        for (int v = 0; v < 8; ++v) {
            float m = rmin[f][v];
            m = fminf(m, __shfl_xor(m, 1, 32));
            m = fminf(m, __shfl_xor(m, 2, 32));
            m = fminf(m, __shfl_xor(m, 4, 32));
            m = fminf(m, __shfl_xor(m, 8, 32));
            s += m;                           // half0: rows f*16+0..7, half1: +8..15
        }
        s += __shfl_xor(s, 16, 32);           // fragment min-sum (all lanes)

        float xs = xn2[f];                    // add back sum of ||x_r||^2 for the fragment
        xs += __shfl_xor(xs, 1, 32);
        xs += __shfl_xor(xs, 2, 32);
        xs += __shfl_xor(xs, 4, 32);
        xs += __shfl_xor(xs, 8, 32);          // each half holds identical row set

        total += s + xs;
    }

    if (lane == 0) atomicAdd(accum, total);
}

__global__ void chamfer_init_ws(float* __restrict__ ws)
{
    if (threadIdx.x < 2) ws[threadIdx.x] = 0.0f;
}

__global__ void chamfer_finalize(const float* __restrict__ ws, float* __restrict__ out,
                                 float invBN, float invBM)
{
    if (threadIdx.x == 0) out[0] = ws[0] * invBN + ws[1] * invBM;
}

extern "C" void kernel_launch(void* const* d_in, const int* in_sizes, int n_in,
                              void* d_out, int out_size, void* d_ws, size_t ws_size,
                              hipStream_t stream)
{
    const float* x = (const float*)d_in[0];   // (B, N, 3) f32
    const float* y = (const float*)d_in[1];   // (B, M, 3) f32
    float* out = (float*)d_out;               // scalar f32
    float* ws  = (float*)d_ws;                // ws[0]: x->y, ws[1]: y->x

    const int B = 4;
    const int N = in_sizes[0] / (B * 3);
    const int M = in_sizes[1] / (B * 3);

    chamfer_init_ws<<<1, 32, 0, stream>>>(ws);

    const int maxRows = (N > M) ? N : M;
    dim3 block(256);
    dim3 grid((maxRows + ROWS_PER_BLOCK - 1) / ROWS_PER_BLOCK, B, 2);
    chamfer_kernel<<<grid, block, 0, stream>>>(x, y, N, M, ws);

    chamfer_finalize<<<1, 32, 0, stream>>>(ws, out,
                                           1.0f / (float)(B * N),
                                           1.0f / (float)(B * M));
}